// CEpsilonLoss_60748017434787
// MI455X (gfx1250) — compile-verified
//
#include <hip/hip_runtime.h>
#include <math.h>

typedef float v2f __attribute__((ext_vector_type(2)));
typedef float v8f __attribute__((ext_vector_type(8)));
typedef __attribute__((address_space(3))) float lds_float;

#define EPSV     0.1f
#define DK       64      // d-chunk staged through LDS (must stay 64: staging uses >>4/&15)
#define MSPLIT   8       // m-range splits across gridDim.y
#define LDSTRIDE 68      // floats per LDS row: 272B, 16B-aligned, bank-conflict-free

// ---------------------------------------------------------------------------
// Kernel 1: fused L1-cdist + streaming log-sum-exp over each block's m-range.
// Block = 256 threads (8 waves). Each wave computes a 16(n) x 16(m) tile.
//  - Tiles staged into LDS with GLOBAL_LOAD_ASYNC_TO_LDS_B128 (ASYNCcnt path).
//  - Per-lane accumulators laid out to match the V_WMMA_F32_16X16X4_F32
//    A-fragment (row n = lane%16; chunk q regs <-> m = 4q + 2*(lane/16)+{0,1}),
//    so the per-tile row reduction of exp() values is 4 chained WMMAs against
//    an all-ones B matrix (layout-proof since B is constant 1.0).
// ---------------------------------------------------------------------------
__global__ __launch_bounds__(256)
void ce_dist_kernel(const float* __restrict__ real,
                    const float* __restrict__ fake,
                    const float* __restrict__ fv,
                    float2* __restrict__ part,
                    int N, int M, int D) {
  __shared__ float  sReal[16 * LDSTRIDE];    // [n][d] row-major, padded
  __shared__ float  sFake[128 * LDSTRIDE];   // [m][d] row-major, padded
  __shared__ float  sFv[256];                // fv for this block's m-range
  __shared__ float2 sRun[8][16];             // per-wave running (max,sum)/row
  __shared__ float  sNewMax[8][16];          // per-tile new max per row

  const int tid    = threadIdx.x;
  const int w      = tid >> 5;               // wave id 0..7
  const int lane   = tid & 31;
  const int n_l    = lane & 15;              // row owned by this lane
  const int h      = lane >> 4;              // half-wave id
  const int n0     = blockIdx.x * 16;
  const int mrange = M / MSPLIT;             // 256 for the reference shapes
  const int mstart = blockIdx.y * mrange;

  for (int i = tid; i < mrange; i += 256) sFv[i] = fv[mstart + i];
  if (lane < 16) sRun[w][lane] = make_float2(-__builtin_inff(), 0.0f);
  __syncthreads();

  const v2f onesB = {1.0f, 1.0f};

  for (int mb = 0; mb < mrange; mb += 128) {   // 128 m-cols per iteration
    const int wt = w * 16;                     // wave's m-offset in the 128
    float2 acc[4];
#pragma unroll
    for (int q = 0; q < 4; ++q) { acc[q].x = 0.0f; acc[q].y = 0.0f; }

    for (int dk = 0; dk < D; dk += DK) {
      // ---- async-stage fake[128][DK]: 2048 16B chunks, 8 per thread ----
      for (int c = tid; c < 128 * (DK / 4); c += 256) {
        const int mr = c >> 4, ch = c & 15;
        lds_float* lp = (lds_float*)&sFake[mr * LDSTRIDE + ch * 4];
        const float* gp = &fake[(size_t)(mstart + mb + mr) * D + dk + ch * 4];
        asm volatile("global_load_async_to_lds_b128 %0, %1, off"
                     :: "v"(lp), "v"(gp) : "memory");
      }
      // ---- async-stage real[16][DK]: 256 chunks, 1 per thread ----
      {
        const int nr = tid >> 4, ch = tid & 15;
        lds_float* lp = (lds_float*)&sReal[nr * LDSTRIDE + ch * 4];
        const float* gp = &real[(size_t)(n0 + nr) * D + dk + ch * 4];
        asm volatile("global_load_async_to_lds_b128 %0, %1, off"
                     :: "v"(lp), "v"(gp) : "memory");
      }
      // prefetch next d-chunk of fake (lowers to global_prefetch_b8)
      if (dk + DK < D) {
        __builtin_prefetch(
            &fake[(size_t)(mstart + mb + (tid >> 1)) * D + dk + DK + ((tid & 1) << 5)],
            0, 0);
      }
      asm volatile("s_wait_asynccnt 0x0" ::: "memory");
      __syncthreads();

      // ---- inner loop: acc += |r - f|, two d per step via float2 ----
#pragma unroll 4
      for (int d = 0; d < DK; d += 2) {
        const float2 r = *reinterpret_cast<const float2*>(&sReal[n_l * LDSTRIDE + d]);
#pragma unroll
        for (int q = 0; q < 4; ++q) {
          const int mA = wt + 4 * q + 2 * h;
          const float2 f0 = *reinterpret_cast<const float2*>(&sFake[mA * LDSTRIDE + d]);
          const float2 f1 = *reinterpret_cast<const float2*>(&sFake[(mA + 1) * LDSTRIDE + d]);
          acc[q].x += fabsf(r.x - f0.x) + fabsf(r.y - f0.y);
          acc[q].y += fabsf(r.x - f1.x) + fabsf(r.y - f1.y);
        }
      }
      __syncthreads();
    }

    // ---- per-tile streaming log-sum-exp update ----
    float2 xv[4];
    float lmax = -__builtin_inff();
#pragma unroll
    for (int q = 0; q < 4; ++q) {
      const int ml = mb + wt + 4 * q + 2 * h;
      const float2 f2 = *reinterpret_cast<const float2*>(&sFv[ml]);
      xv[q].x = (f2.x - acc[q].x) * EPSV;
      xv[q].y = (f2.y - acc[q].y) * EPSV;
      lmax = fmaxf(lmax, fmaxf(xv[q].x, xv[q].y));
    }
    const float tmax = fmaxf(lmax, __shfl_xor(lmax, 16, 32));
    const float rmax_old = sRun[w][n_l].x;
    const float nm = fmaxf(rmax_old, tmax);
    if (h == 0) sNewMax[w][n_l] = nm;

    // exp() in A-fragment layout, then 4 chained WMMAs sum the 16 m per row.
    v8f csum = {};
#pragma unroll
    for (int q = 0; q < 4; ++q) {
      v2f a;
      a.x = __expf(xv[q].x - nm);
      a.y = __expf(xv[q].y - nm);
      csum = __builtin_amdgcn_wmma_f32_16x16x4_f32(
          /*neg_a=*/false, a, /*neg_b=*/false, onesB,
          /*c_mod=*/(short)0, csum, /*reuse_a=*/false, /*reuse_b=*/false);
    }
    __syncthreads();

    // lanes 0 and 16 hold the row sums (rows 0-7 / 8-15) in csum[0..7]
    if (n_l == 0) {
#pragma unroll
      for (int v = 0; v < 8; ++v) {
        const int row = h * 8 + v;
        float2 run = sRun[w][row];
        const float nmr = sNewMax[w][row];
        run.y = run.y * __expf(run.x - nmr) + csum[v];
        run.x = nmr;
        sRun[w][row] = run;
      }
    }
    __syncthreads();
  }

  // ---- merge the 8 wave partials per row, emit (max,sum) to workspace ----
  if (tid < 16) {
    float mx = -__builtin_inff(), s = 0.0f;
#pragma unroll
    for (int ww = 0; ww < 8; ++ww) {
      const float2 r = sRun[ww][tid];
      const float nm2 = fmaxf(mx, r.x);
      s = s * __expf(mx - nm2) + r.y * __expf(r.x - nm2);
      mx = nm2;
    }
    part[(size_t)blockIdx.y * N + n0 + tid] = make_float2(mx, s);
  }
}

// ---------------------------------------------------------------------------
// Kernel 2: merge MSPLIT partials per row -> term2[n]; reduce to the scalar.
// ---------------------------------------------------------------------------
__global__ __launch_bounds__(256)
void ce_final_kernel(const float2* __restrict__ part,
                     const float* __restrict__ fv,
                     float* __restrict__ out, int N, int M) {
  __shared__ float rA[256];
  __shared__ float rB[256];
  const int tid = threadIdx.x;

  float tsum = 0.0f;
  for (int n = tid; n < N; n += 256) {
    float mx = -__builtin_inff(), s = 0.0f;
#pragma unroll
    for (int p = 0; p < MSPLIT; ++p) {
      const float2 r = part[(size_t)p * N + n];
      const float nm = fmaxf(mx, r.x);
      s = s * __expf(mx - nm) + r.y * __expf(r.x - nm);
      mx = nm;
    }
    tsum += mx + __logf(s) - __logf((float)M);   // log(mean_m exp(x))
  }
  float fsum = 0.0f;
  for (int m = tid; m < M; m += 256) fsum += fv[m];

  rA[tid] = tsum;
  rB[tid] = fsum;
  __syncthreads();
  for (int st = 128; st > 0; st >>= 1) {
    if (tid < st) { rA[tid] += rA[tid + st]; rB[tid] += rB[tid + st]; }
    __syncthreads();
  }
  if (tid == 0)
    out[0] = -(rB[0] / (float)M) + (rA[0] / (float)N) / EPSV;
}

// ---------------------------------------------------------------------------
extern "C" void kernel_launch(void* const* d_in, const int* in_sizes, int n_in,
                              void* d_out, int out_size, void* d_ws, size_t ws_size,
                              hipStream_t stream) {
  const float* real = (const float*)d_in[0];   // [N, D] f32
  const float* fake = (const float*)d_in[1];   // [M, D] f32
  const float* fv   = (const float*)d_in[2];   // [M]    f32

  const int M = in_sizes[2];
  const int D = in_sizes[1] / M;
  const int N = in_sizes[0] / D;

  float2* part = (float2*)d_ws;                // [MSPLIT][N] (max,sum) pairs

  dim3 grid(N / 16, MSPLIT);
  ce_dist_kernel<<<grid, 256, 0, stream>>>(real, fake, fv, part, N, M, D);
  ce_final_kernel<<<1, 256, 0, stream>>>(part, fv, (float*)d_out, N, M);
}